// GraphLayer_8512625180826
// MI455X (gfx1250) — compile-verified
//
#include <hip/hip_runtime.h>

// MI455X / gfx1250, wave32. Distances + conv via V_WMMA_F32_16X16X4_F32
// (fp32 matrix path -> kNN ranking matches the fp32 reference). Candidate
// tiles staged block-wide into LDS with CDNA5 async loads
// (GLOBAL_LOAD_ASYNC_TO_LDS_B128, ASYNCcnt) with double buffering.

typedef __attribute__((ext_vector_type(2))) float v2f;
typedef __attribute__((ext_vector_type(4))) float v4f;
typedef __attribute__((ext_vector_type(8))) float v8f;

#define NB     8
#define NPTS   4096
#define CIN    64
#define COUT   128
#define KNN    16
#define ROWS   (NB * NPTS)
#define BIGF   3.0e38f

#define STG_CANDS 64              // candidates staged per block per stage
#define STG_PITCH 68              // row pitch in floats: 272B = 16B aligned, bank-spread
#define NSTAGE    (NPTS / STG_CANDS)

// ---------------------------------------------------------------- stats init
__global__ void k_zero_stats(float* stats) {
    if (threadIdx.x < 2 * COUT) stats[threadIdx.x] = 0.0f;
}

// ---------------------------------------------------------------- sq norms
__global__ void k_sqnorm(const float* __restrict__ x, float* __restrict__ sq) {
    int row = blockIdx.x * blockDim.x + threadIdx.x;
    if (row >= ROWS) return;
    const float* p = x + (size_t)row * CIN;
    float s = 0.0f;
#pragma unroll
    for (int c = 0; c < CIN; c += 4) {
        v4f v = *(const v4f*)(p + c);
        s += v.x * v.x + v.y * v.y + v.z * v.z + v.w * v.w;
    }
    sq[row] = s;
}

// ---------------------------------------------------------------- kNN + gather + maxpool
// One block (8 waves) = 128 consecutive queries of one batch; each wave owns
// 16 queries. All waves share LDS-staged candidate tiles (64 cands x 64 ch),
// double-buffered via async global->LDS copies. Per 16-cand chunk a wave does
// 16 chained V_WMMA_F32_16X16X4_F32 (K=64). D layout: lane<16 -> M=0..7,
// N=lane; lane>=16 -> M=8..15, N=lane-16 -> each lane streams 8 candidates
// for its query into a register top-16 (replace-max), lane pairs merge in LDS,
// then fused neighbor gather + channel maxpool.
__global__ __launch_bounds__(256)
void k_knn_pool(const float* __restrict__ x, const float* __restrict__ sq,
                float* __restrict__ pooled) {
    __shared__ union {
        float stage[2][STG_CANDS * STG_PITCH];            // 34816 B, candidate loop
        struct {                                          // 40960 B, merge phase
            float d[8][512];
            int   i[8][512];
            int   sel[8][16][KNN];
        } m;
    } sh __attribute__((aligned(16)));

    const int lane = threadIdx.x & 31;
    const int wave = threadIdx.x >> 5;
    const int b      = blockIdx.x >> 5;                   // 32 blocks per batch
    const int qblock = (blockIdx.x & 31) << 7;            // 128 queries per block
    const int qbase  = qblock + wave * 16;
    const float* xb  = x  + (size_t)b * NPTS * CIN;
    const float* sqb = sq + (size_t)b * NPTS;

    const int l15 = lane & 15;
    const int hi  = lane >> 4;

    // issue one 64-cand x 64-ch stage as 4 async B128 copies per thread
    auto issue_stage = [&](int s, int bufsel) {
        const float* gp = xb + (size_t)s * STG_CANDS * CIN;
        const int t = threadIdx.x;
#pragma unroll
        for (int q = 0; q < 4; ++q) {
            int i    = t + q * 256;                       // float4 index 0..1023
            int cand = i >> 4;
            int ch4  = (i & 15) << 2;
            unsigned loff = (unsigned)(unsigned long long)
                            &sh.stage[bufsel][cand * STG_PITCH + ch4];
            const float* ga = gp + cand * CIN + ch4;
            asm volatile("global_load_async_to_lds_b128 %0, %1, off"
                         :: "v"(loff), "v"(ga) : "memory");
        }
    };

    // Preload B-matrix tiles (this wave's 16 queries), one v2f per K-step of 4:
    // B[k][n]: lane<16 holds K=4t,4t+1 for N=lane; lane>=16 holds K=4t+2,4t+3.
    v2f bt[16];
#pragma unroll
    for (int t = 0; t < 16; ++t)
        bt[t] = *(const v2f*)(xb + (size_t)(qbase + l15) * CIN + 4 * t + 2 * hi);

    // per-lane top-16 (unsorted; replace-max scheme, fully unrolled)
    float kd[16]; int ki[16];
#pragma unroll
    for (int j = 0; j < 16; ++j) { kd[j] = BIGF; ki[j] = 0; }
    float curmax = BIGF; int maxpos = 0;

    issue_stage(0, 0);
    for (int s = 0; s < NSTAGE; ++s) {
        const int cur = s & 1;
        if (s + 1 < NSTAGE) {
            issue_stage(s + 1, cur ^ 1);
            asm volatile("s_wait_asynccnt 0x4" ::: "memory");   // stage s arrived
        } else {
            asm volatile("s_wait_asynccnt 0x0" ::: "memory");
        }
        __syncthreads();                                        // all waves' data visible

        const float* sb = sh.stage[cur];
#pragma unroll
        for (int chunk = 0; chunk < STG_CANDS / 16; ++chunk) {
            const int cl = chunk * 16 + l15;                    // local candidate row
            v8f acc = {};
#pragma unroll
            for (int t = 0; t < 16; ++t) {
                // A[m][k]: lane<16 K=4t,4t+1 (M=lane); lane>=16 K=4t+2,4t+3
                v2f at = *(const v2f*)(sb + cl * STG_PITCH + 4 * t + 2 * hi);
                acc = __builtin_amdgcn_wmma_f32_16x16x4_f32(
                    false, at, false, bt[t], (short)0, acc, false, false);
            }
            const int mbase = s * STG_CANDS + chunk * 16 + hi * 8;
            v4f sq0 = *(const v4f*)(sqb + mbase);
            v4f sq1 = *(const v4f*)(sqb + mbase + 4);
            float sv[8] = {sq0.x, sq0.y, sq0.z, sq0.w, sq1.x, sq1.y, sq1.z, sq1.w};
#pragma unroll
            for (int r = 0; r < 8; ++r) {
                // rank value: ||xj||^2 - 2<xi,xj>  (query norm is rank-invariant)
                float v  = sv[r] - 2.0f * acc[r];
                int   id = mbase + r;
                if (v < curmax) {
#pragma unroll
                    for (int j = 0; j < 16; ++j)
                        if (j == maxpos) { kd[j] = v; ki[j] = id; }
                    float m = -BIGF; int mp = 0;
#pragma unroll
                    for (int j = 0; j < 16; ++j)
                        if (kd[j] > m) { m = kd[j]; mp = j; }
                    curmax = m; maxpos = mp;
                }
            }
        }
        __syncthreads();                        // done reading buf before re-fill
    }

    // dump per-lane lists, merge lane l with lane l+16 (same query)
#pragma unroll
    for (int j = 0; j < 16; ++j) {
        sh.m.d[wave][lane * 16 + j] = kd[j];
        sh.m.i[wave][lane * 16 + j] = ki[j];
    }
    __syncthreads();
    if (lane < 16) {
        for (int sel = 0; sel < KNN; ++sel) {
            float m = BIGF; int mpos = lane * 16;
#pragma unroll
            for (int j = 0; j < 16; ++j) {
                float v0 = sh.m.d[wave][lane * 16 + j];
                if (v0 < m) { m = v0; mpos = lane * 16 + j; }
                float v1 = sh.m.d[wave][(lane + 16) * 16 + j];
                if (v1 < m) { m = v1; mpos = (lane + 16) * 16 + j; }
            }
            sh.m.sel[wave][lane][sel] = sh.m.i[wave][mpos];
            sh.m.d[wave][mpos] = BIGF;          // remove from pool
        }
    }
    __syncthreads();

    // gather + maxpool: lane handles query l15, channel half hi (32 channels)
    v4f mx[8];
#pragma unroll
    for (int c = 0; c < 8; ++c) mx[c] = (v4f){-BIGF, -BIGF, -BIGF, -BIGF};
    for (int k = 0; k < KNN; ++k) {
        int nb = sh.m.sel[wave][l15][k];
        const float* px = xb + (size_t)nb * CIN + hi * 32;
#pragma unroll
        for (int c = 0; c < 8; ++c) {
            v4f v = *(const v4f*)(px + 4 * c);
            mx[c].x = fmaxf(mx[c].x, v.x);
            mx[c].y = fmaxf(mx[c].y, v.y);
            mx[c].z = fmaxf(mx[c].z, v.z);
            mx[c].w = fmaxf(mx[c].w, v.w);
        }
    }
    float* pout = pooled + (size_t)(b * NPTS + qbase + l15) * CIN + hi * 32;
#pragma unroll
    for (int c = 0; c < 8; ++c) *(v4f*)(pout + 4 * c) = mx[c];
}

// ---------------------------------------------------------------- conv1x1 + BN stats
// One wave = 16 rows of pooled x all 128 outputs. 8 accumulator tiles, K=64
// over 16 WMMA steps each. Lane owns channel o = 16*j + (lane&15) and rows
// rbase + (lane>>4)*8 + r -> per-lane partial sum/sumsq, global f32 atomics.
__global__ __launch_bounds__(256)
void k_conv_stats(const float* __restrict__ pooled, const float* __restrict__ W,
                  const float* __restrict__ bias, float* __restrict__ y,
                  float* __restrict__ stats) {
    const int lane  = threadIdx.x & 31;
    const int wave  = threadIdx.x >> 5;
    const int rbase = (blockIdx.x * 8 + wave) * 16;  // 2048 row-tiles
    const int l15 = lane & 15;
    const int hi  = lane >> 4;

    v8f acc[8];
#pragma unroll
    for (int j = 0; j < 8; ++j) acc[j] = (v8f){};

#pragma unroll
    for (int t = 0; t < 16; ++t) {
        v2f at = *(const v2f*)(pooled + (size_t)(rbase + l15) * CIN + 4 * t + 2 * hi);
#pragma unroll
        for (int j = 0; j < 8; ++j) {
            // B[k][n] = W[16j + n][k]  (y = pooled x W^T)
            v2f bw = *(const v2f*)(W + (size_t)(j * 16 + l15) * CIN + 4 * t + 2 * hi);
            acc[j] = __builtin_amdgcn_wmma_f32_16x16x4_f32(
                false, at, false, bw, (short)0, acc[j], false, false);
        }
    }

#pragma unroll
    for (int j = 0; j < 8; ++j) {
        int   o  = j * 16 + l15;
        float bb = bias[o];
        float s = 0.0f, s2 = 0.0f;
#pragma unroll
        for (int r = 0; r < 8; ++r) {
            float v   = acc[j][r] + bb;
            int   row = rbase + hi * 8 + r;
            y[(size_t)row * COUT + o] = v;
            s += v; s2 += v * v;
        }
        atomicAdd(&stats[o], s);
        atomicAdd(&stats[COUT + o], s2);
    }
}

// ---------------------------------------------------------------- BN finalize + ReLU
__global__ void k_bn_relu(const float* __restrict__ y, const float* __restrict__ stats,
                          const float* __restrict__ gamma, const float* __restrict__ beta,
                          float* __restrict__ out) {
    int i = blockIdx.x * blockDim.x + threadIdx.x;
    if (i >= ROWS * COUT) return;
    int o = i & (COUT - 1);
    const float inv = 1.0f / (float)ROWS;
    float mean = stats[o] * inv;
    float var  = stats[COUT + o] * inv - mean * mean;   // population var (jnp.var)
    float sc   = gamma[o] * rsqrtf(var + 1e-5f);
    float v    = (y[i] - mean) * sc + beta[o];
    out[i] = fmaxf(v, 0.0f);
}

// ---------------------------------------------------------------- launcher
extern "C" void kernel_launch(void* const* d_in, const int* in_sizes, int n_in,
                              void* d_out, int out_size, void* d_ws, size_t ws_size,
                              hipStream_t stream) {
    const float* x     = (const float*)d_in[0];   // [8,4096,64]
    const float* W     = (const float*)d_in[1];   // [128,64]
    const float* bias  = (const float*)d_in[2];   // [128]
    const float* gamma = (const float*)d_in[3];   // [128]
    const float* beta  = (const float*)d_in[4];   // [128]
    float* out = (float*)d_out;                   // [8,4096,128]

    // workspace layout (floats): stats[256] | sq[32768] | pooled[ROWS*64] | y[ROWS*128]
    float* stats  = (float*)d_ws;
    float* sq     = stats + 2 * COUT;
    float* pooled = sq + ROWS;
    float* y      = pooled + (size_t)ROWS * CIN;

    k_zero_stats<<<1, 256, 0, stream>>>(stats);
    k_sqnorm<<<ROWS / 256, 256, 0, stream>>>(x, sq);
    k_knn_pool<<<256, 256, 0, stream>>>(x, sq, pooled);       // 2048 waves
    k_conv_stats<<<256, 256, 0, stream>>>(pooled, W, bias, y, stats);
    k_bn_relu<<<(ROWS * COUT) / 256, 256, 0, stream>>>(y, stats, gamma, beta, out);
}